// UVFeatureFusion_82600811037116
// MI455X (gfx1250) — compile-verified
//
#include <hip/hip_runtime.h>
#include <cstdint>
#include <cstddef>

// Problem constants (fixed by the reference: B=2, V=6, C=64, H=W=A=256)
#define HW_     65536        // H*W
#define VHW_    393216       // V*H*W
#define NPIX_   786432       // B*V*H*W
#define CCH_    64           // channels
#define AA_     65536        // A*A
#define BB_     2
#define NFUSED_ (BB_ * CCH_ * AA_)   // 8388608 fused output elements

// ---------------------------------------------------------------------------
// Kernel 1: per-pixel atlas index + validity; scatter counts.
// gidx_out[n] = b*A*A + y*A + x  (or -1 if invalid)
// ---------------------------------------------------------------------------
__global__ __launch_bounds__(256)
void uv_index_kernel(const float* __restrict__ uv,
                     const float* __restrict__ mask,
                     int* __restrict__ gidx_out,
                     float* __restrict__ counts) {
    int n = blockIdx.x * 256 + threadIdx.x;
    float u = uv[2 * n + 0];
    float v = uv[2 * n + 1];
    float m = mask[n];

    bool fin   = __builtin_isfinite(u) && __builtin_isfinite(v);
    bool valid = (m > 0.5f) && fin;

    // clip to [0,1]; maxNum/minNum semantics map NaN -> 0 (matches clip+nan_to_num,
    // and the weight is 0 for non-finite uv anyway)
    float uc = fminf(fmaxf(u, 0.0f), 1.0f);
    float vc = fminf(fmaxf(v, 0.0f), 1.0f);

    int x = (int)rintf(uc * 255.0f);            // round-half-even == jnp.round
    int y = (int)rintf((1.0f - vc) * 255.0f);
    int b = (n >= VHW_) ? 1 : 0;                // B == 2
    int g = (b << 16) | (y << 8) | x;           // b*A*A + y*A + x

    gidx_out[n] = valid ? g : -1;
    if (valid) atomicAdd(&counts[g], 1.0f);
}

// ---------------------------------------------------------------------------
// Kernel 2: scatter features into the fused atlas ([B,C,A,A] layout).
// Grid: (NPIX/1024, C). Each block: 1024 consecutive pixels of one channel.
// Features are staged global -> LDS with the CDNA5 async-copy path
// (global_load_async_to_lds_b128 / s_wait_asynccnt), then scattered with
// global f32 atomics (destination is L2-resident: 34 MB << 192 MB L2).
// ---------------------------------------------------------------------------
__global__ __launch_bounds__(256)
void uv_scatter_kernel(const float* __restrict__ feats,
                       const int* __restrict__ gidx_in,
                       float* __restrict__ fused) {
    __shared__ float smem[1024];

    int t  = threadIdx.x;
    int c  = blockIdx.y;
    int n0 = blockIdx.x * 1024;          // tile base pixel; 1024 | 65536 so the
    int bv = n0 >> 16;                   // tile never straddles a (b,v) plane
    int pin = n0 & (HW_ - 1);

    // feats flat index for (bv, c, pixel-in-plane): (bv*C + c)*HW + pin
    const float* gp = feats + (((size_t)(bv * CCH_ + c)) << 16) + (size_t)pin + 4u * t;

    // LDS generic pointers on AMDGPU are {SHARED_BASE_HI, lds_offset}: the low
    // 32 bits ARE the LDS byte offset needed by the async-load VDST operand.
    unsigned lds_off = (unsigned)(uintptr_t)(&smem[4 * t]);

    // 16 B per lane, coalesced 512 B/wave, async into LDS (tracked by ASYNCcnt)
    asm volatile("global_load_async_to_lds_b128 %0, %1, off"
                 :: "v"(lds_off), "v"(gp)
                 : "memory");

    // Overlap: fetch the 4 cell indices while the async copy is in flight.
    int4 g4 = ((const int4*)(gidx_in + n0))[t];

    asm volatile("s_wait_asynccnt 0" ::: "memory");
    // Each lane reads back only its own 16 B -> no cross-lane barrier needed.
    float4 f = *(const float4*)(&smem[4 * t]);

    int   gs[4] = {g4.x, g4.y, g4.z, g4.w};
    float fv[4] = {f.x, f.y, f.z, f.w};
#pragma unroll
    for (int k = 0; k < 4; ++k) {
        int g = gs[k];
        if (g >= 0) {
            int b = g >> 16;
            size_t oidx = (((size_t)(b * CCH_ + c)) << 16) | (unsigned)(g & 0xFFFF);
            atomicAdd(&fused[oidx], fv[k]);
        }
    }
}

// ---------------------------------------------------------------------------
// Kernel 3: fused /= max(counts,1); validity = counts > 0.
// One thread per fused element; each count is read 64x but stays in L2/L0.
// ---------------------------------------------------------------------------
__global__ __launch_bounds__(256)
void uv_normalize_kernel(float* __restrict__ fused,
                         const float* __restrict__ counts,
                         float* __restrict__ validity) {
    int idx  = blockIdx.x * 256 + threadIdx.x;   // [0, B*C*A*A)
    int b    = idx >> 22;                        // C*A*A = 2^22
    int c    = (idx >> 16) & (CCH_ - 1);
    int cell = idx & 0xFFFF;
    int gi   = (b << 16) | cell;

    float cnt = counts[gi];
    fused[idx] = fused[idx] * (1.0f / fmaxf(cnt, 1.0f));
    if (c == 0) validity[gi] = (cnt > 0.0f) ? 1.0f : 0.0f;
}

// ---------------------------------------------------------------------------
// Launch: memsets + 3 kernels, all on `stream` (graph-capture safe).
// Workspace layout: [0, 512KB) counts (B*A*A f32); [512KB, 3.5MB) gidx (N i32)
// ---------------------------------------------------------------------------
extern "C" void kernel_launch(void* const* d_in, const int* in_sizes, int n_in,
                              void* d_out, int out_size, void* d_ws, size_t ws_size,
                              hipStream_t stream) {
    (void)in_sizes; (void)n_in; (void)out_size; (void)ws_size;

    const float* feats = (const float*)d_in[0];  // [B,V,C,H,W] f32
    const float* uv    = (const float*)d_in[1];  // [B,V,H,W,2] f32
    const float* mask  = (const float*)d_in[2];  // [B,V,1,H,W] f32
    // d_in[3] = atlas_size (256), fixed by the reference shapes.

    float* fused    = (float*)d_out;             // [B,C,A,A]
    float* validity = fused + NFUSED_;           // [B,1,A,A]

    float* counts = (float*)d_ws;                                   // B*A*A f32
    int*   gidx   = (int*)((char*)d_ws + (size_t)BB_ * AA_ * sizeof(float));

    // Deterministic accumulator init every call (capturable memset nodes).
    hipMemsetAsync(fused, 0, (size_t)NFUSED_ * sizeof(float), stream);
    hipMemsetAsync(counts, 0, (size_t)BB_ * AA_ * sizeof(float), stream);

    uv_index_kernel<<<NPIX_ / 256, 256, 0, stream>>>(uv, mask, gidx, counts);

    dim3 sg(NPIX_ / 1024, CCH_);
    uv_scatter_kernel<<<sg, 256, 0, stream>>>(feats, gidx, fused);

    uv_normalize_kernel<<<NFUSED_ / 256, 256, 0, stream>>>(fused, counts, validity);
}